// DigitCapsule_91036126806238
// MI455X (gfx1250) — compile-verified
//
#include <hip/hip_runtime.h>
#include <math.h>

// Problem constants (B, R, O, DO, DI) = (128, 1152, 10, 16, 8)
#define BB   128
#define RR   1152
#define OO   10
#define DOUT 16
#define DIN  8
#define MDIM (OO * DOUT)   // 160 = flattened (o,k)

typedef __attribute__((ext_vector_type(2))) float v2f;
typedef __attribute__((ext_vector_type(4))) float v4f;
typedef __attribute__((ext_vector_type(8))) float v8f;

// -----------------------------------------------------------------------------
// Kernel 1: u_hat[b][r][m] = sum_i W[r][m][i] * x[b][r][i]   (m = o*16 + k)
// Per r: GEMM (M=160) x (K=8) x (N=128) done with V_WMMA_F32_16X16X4_F32.
// One block per r, 8 waves; wave w owns batch-tile w (N = w*16..w*16+15) and
// loops over the 10 m-tiles. K=8 -> two chained 16x16x4 WMMAs per tile.
// -----------------------------------------------------------------------------
__global__ __launch_bounds__(256) void uhat_wmma_kernel(
    const float* __restrict__ x,     // [B][R][DI]
    const float* __restrict__ Wg,    // [R][O][DO][DI] = [R][160][8]
    float* __restrict__ uhat)        // [B][R][160]
{
    __shared__ float Wlds[MDIM][9];  // padded stride 9 -> conflict-free column reads
    __shared__ float xlds[DIN][BB];  // x transposed per-r: [i][b]

    const int r = blockIdx.x;
    const int t = threadIdx.x;

    // Stage W[r] (160x8, 5KB) and x[:,r,:]^T (8x128, 4KB) into LDS.
    const float* Wr = Wg + (size_t)r * MDIM * DIN;
    for (int idx = t; idx < MDIM * DIN; idx += 256)
        Wlds[idx >> 3][idx & 7] = Wr[idx];
    for (int idx = t; idx < DIN * BB; idx += 256) {
        int b = idx >> 3, i = idx & 7;
        xlds[i][b] = x[((size_t)b * RR + r) * DIN + i];
    }
    __syncthreads();

    const int lane = t & 31;
    const int wave = t >> 5;       // n-tile (batch tile) 0..7
    const int half = lane >> 4;    // 0: lanes 0-15, 1: lanes 16-31
    const int l16  = lane & 15;
    const int bcol = wave * 16 + l16;   // batch index handled by this lane (N)

    // B fragment (4x16 f32): lanes 0-15 hold {K+0 in v0, K+1 in v1},
    // lanes 16-31 hold {K+2, K+3}. vb0 covers K=0..3, vb1 covers K=4..7.
    v2f vb0, vb1;
    vb0.x = xlds[2 * half + 0][bcol];
    vb0.y = xlds[2 * half + 1][bcol];
    vb1.x = xlds[4 + 2 * half + 0][bcol];
    vb1.y = xlds[4 + 2 * half + 1][bcol];

    for (int mt = 0; mt < 10; ++mt) {
        const int M = mt * 16 + l16;   // row of A handled by this lane
        // A fragment (16x4 f32): lanes 0-15 {K0,K1}, lanes 16-31 {K2,K3}.
        v2f va0, va1;
        va0.x = Wlds[M][2 * half + 0];
        va0.y = Wlds[M][2 * half + 1];
        va1.x = Wlds[M][4 + 2 * half + 0];
        va1.y = Wlds[M][4 + 2 * half + 1];

        v8f acc = {0.f, 0.f, 0.f, 0.f, 0.f, 0.f, 0.f, 0.f};
        acc = __builtin_amdgcn_wmma_f32_16x16x4_f32(false, va0, false, vb0,
                                                    (short)0, acc, false, false);
        acc = __builtin_amdgcn_wmma_f32_16x16x4_f32(false, va1, false, vb1,
                                                    (short)0, acc, false, false);

        // D layout: lane = column N (=batch), v0..v7 = rows M = mt*16 + half*8 + j.
        // -> 8 contiguous floats of uhat[b][r][.], store as two b128s.
        float* dst = uhat + ((size_t)bcol * RR + r) * MDIM + mt * 16 + half * 8;
        v4f lo = {acc[0], acc[1], acc[2], acc[3]};
        v4f hi = {acc[4], acc[5], acc[6], acc[7]};
        *(v4f*)(dst)     = lo;
        *(v4f*)(dst + 4) = hi;
    }
}

// -----------------------------------------------------------------------------
// Kernel 2: fused 3-iteration dynamic routing. One block per batch element b.
// Logits b_ij[r][o] live in LDS (45KB). 320 threads = 10 waves (wave o does the
// softmax stats for output capsule o via wave32 shuffles).
// -----------------------------------------------------------------------------
__global__ __launch_bounds__(320) void routing_kernel(
    const float* __restrict__ uhat,  // [B][R][160]
    float* __restrict__ out)         // [B][O][DO][1] = [B][160]
{
    __shared__ float lg[RR * OO];       // routing logits b_ij, layout [r][o]
    __shared__ float spart[2][MDIM];    // partial s over r-halves
    __shared__ float vv[MDIM];          // squashed v_j
    __shared__ float mxs[OO];           // softmax max per o
    __shared__ float dns[OO];           // softmax denom per o

    const int b = blockIdx.x;
    const int t = threadIdx.x;
    const float* U = uhat + (size_t)b * RR * MDIM;

    for (int idx = t; idx < RR * OO; idx += 320) lg[idx] = 0.f;
    __syncthreads();

    const int lane = t & 31;
    const int wave = t >> 5;        // 0..9 -> output capsule o for softmax stats
    const int m    = t % MDIM;      // (o,k) column, 0..159
    const int seg  = t / MDIM;      // 0 or 1: which half of R this thread sums
    const int om   = m / DOUT;      // o for this column

    for (int iter = 0; iter < 3; ++iter) {
        // ---- softmax stats over r for each o (matches jax.nn.softmax axis=1)
        {
            const int o = wave;
            float mx = -INFINITY;
            for (int r = lane; r < RR; r += 32)
                mx = fmaxf(mx, lg[r * OO + o]);
            for (int off = 16; off > 0; off >>= 1)
                mx = fmaxf(mx, __shfl_xor(mx, off, 32));
            float sm = 0.f;
            for (int r = lane; r < RR; r += 32)
                sm += __expf(lg[r * OO + o] - mx);
            for (int off = 16; off > 0; off >>= 1)
                sm += __shfl_xor(sm, off, 32);
            if (lane == 0) { mxs[o] = mx; dns[o] = sm; }
        }
        __syncthreads();

        // ---- s[m] = (sum_r exp(lg[r][o]-mx) * u_hat[r][m]) / denom[o]
        {
            const float mxo = mxs[om];
            float acc = 0.f;
            const int r0 = seg * (RR / 2), r1 = r0 + (RR / 2);
            #pragma unroll 4
            for (int r = r0; r < r1; ++r)
                acc += __expf(lg[r * OO + om] - mxo) * U[(size_t)r * MDIM + m];
            spart[seg][m] = acc;
        }
        __syncthreads();

        // ---- elementwise squash (reference squashes over a size-1 axis)
        if (t < MDIM) {
            float s  = (spart[0][t] + spart[1][t]) / dns[t / DOUT];
            float s2 = s * s;
            float v  = (s2 / (1.f + s2)) * s * rsqrtf(s2 + 1e-9f);
            vv[t] = v;
            if (iter == 2) out[(size_t)b * MDIM + t] = v;
        }
        __syncthreads();

        // ---- logit update: b_ij[r][o] += sum_k u_hat[r][o][k] * v[o][k]
        if (iter < 2) {
            for (int idx = t; idx < RR * OO; idx += 320) {
                const int r = idx / OO, o = idx % OO;
                const float* u = U + (size_t)r * MDIM + o * DOUT;
                float accv = 0.f;
                #pragma unroll
                for (int k = 0; k < DOUT; ++k)
                    accv += u[k] * vv[o * DOUT + k];
                lg[idx] += accv;
            }
            __syncthreads();
        }
    }
}

// -----------------------------------------------------------------------------
extern "C" void kernel_launch(void* const* d_in, const int* in_sizes, int n_in,
                              void* d_out, int out_size, void* d_ws, size_t ws_size,
                              hipStream_t stream) {
    (void)in_sizes; (void)n_in; (void)out_size; (void)ws_size;
    const float* x  = (const float*)d_in[0];   // (128, 1152, 8) f32
    const float* Wg = (const float*)d_in[1];   // (1, 1152, 10, 16, 8) f32
    float* out  = (float*)d_out;               // (128, 10, 16, 1) f32
    float* uhat = (float*)d_ws;                // 128*1152*160*4 = 94.4 MB scratch

    uhat_wmma_kernel<<<RR, 256, 0, stream>>>(x, Wg, uhat);
    routing_kernel<<<BB, 320, 0, stream>>>(uhat, out);
}